// AKDNLayer_50775103373667
// MI455X (gfx1250) — compile-verified
//
#include <hip/hip_runtime.h>
#include <hip/hip_bf16.h>

#define D 64

typedef float v2f __attribute__((ext_vector_type(2)));
typedef float v8f __attribute__((ext_vector_type(8)));

__device__ __forceinline__ v8f wmma4(v2f a, v2f b, v8f c) {
  // D = A(16x4 f32) * B(4x16 f32) + C(16x16 f32), full f32 precision
  return __builtin_amdgcn_wmma_f32_16x16x4_f32(false, a, false, b, (short)0, c,
                                               false, false);
}

// order-preserving float -> uint key for atomicMax-based segment max
__device__ __forceinline__ unsigned fkey(float f) {
  unsigned b = __float_as_uint(f);
  return b ^ ((b & 0x80000000u) ? 0xFFFFFFFFu : 0x80000000u);
}
__device__ __forceinline__ float keyf(unsigned k) {
  unsigned b = (k & 0x80000000u) ? (k ^ 0x80000000u) : ~k;
  return __uint_as_float(b);
}

__global__ void zero_f32_kernel(float* __restrict__ p, long n) {
  long i = (long)blockIdx.x * blockDim.x + threadIdx.x;
  long stride = (long)gridDim.x * blockDim.x;
  for (; i < n; i += stride) p[i] = 0.0f;
}

__global__ void copy_f32v4_kernel(const float* __restrict__ src,
                                  float* __restrict__ dst, long n4) {
  long i = (long)blockIdx.x * blockDim.x + threadIdx.x;
  long stride = (long)gridDim.x * blockDim.x;
  const float4* s4 = (const float4*)src;
  float4* d4 = (float4*)dst;
  for (; i < n4; i += stride) d4[i] = s4[i];
}

__global__ void wkeff_kernel(const float* __restrict__ wkw,
                             float* __restrict__ wkeff) {
  int i = blockIdx.x * blockDim.x + threadIdx.x;
  if (i < D * D) {
    int r = i >> 6, c = i & 63;
    wkeff[i] = wkw[r * 2 * D + c] + wkw[r * 2 * D + D + c];
  }
}

// ---------------------------------------------------------------------------
// Phase 1: att[e] = leakyrelu( <r_e, Wk_eff @ (v_e * i_e) + b> ), fused with
// segment-max (atomicMax on ordered keys). One wave32 per 16-edge tile,
// f32 WMMA 16x16x4, Wk_eff staged in LDS.
// ---------------------------------------------------------------------------
__global__ void __launch_bounds__(256) kg_att_kernel(
    const float* __restrict__ item_emb, const float* __restrict__ entity_emb,
    const float* __restrict__ relation_emb, const float* __restrict__ wkeff,
    const float* __restrict__ wkb, const int* __restrict__ kg_item,
    const int* __restrict__ kg_rel, const int* __restrict__ kg_ent,
    float* __restrict__ att, unsigned* __restrict__ segkey, int nkg,
    int ntiles) {
  __shared__ float sW[D * D];
  __shared__ float sB[D];
  for (int i = threadIdx.x; i < D * D; i += blockDim.x) sW[i] = wkeff[i];
  if (threadIdx.x < D) sB[threadIdx.x] = wkb[threadIdx.x];
  __syncthreads();

  const int lane = threadIdx.x & 31;
  const int wave = threadIdx.x >> 5;
  const int l15 = lane & 15;
  const int h = lane >> 4;

  const int t = blockIdx.x * 8 + wave;
  if (t >= ntiles) return;

  // A-matrix row for this lane: m = l15; K split across lane halves:
  // VGPR pair holds (k0+2h, k0+2h+1) for k-step k0=4*kk
  int rowm = t * 16 + l15;
  if (rowm >= nkg) rowm = nkg - 1;
  const int ii = kg_item[rowm];
  const int ie = kg_ent[rowm];
  float2 hv[16];
#pragma unroll
  for (int kk = 0; kk < 16; ++kk) {
    int k = 4 * kk + 2 * h;
    float2 iv = *(const float2*)(item_emb + (size_t)ii * D + k);
    float2 ev = *(const float2*)(entity_emb + (size_t)ie * D + k);
    float2 p;
    p.x = iv.x * ev.x;
    p.y = iv.y * ev.y;
    hv[kk] = p;
  }

  v8f acc0 = {}, acc1 = {}, acc2 = {}, acc3 = {};
#pragma unroll
  for (int kk = 0; kk < 16; ++kk) {
    int k = 4 * kk + 2 * h;
    v2f a;
    a.x = hv[kk].x;
    a.y = hv[kk].y;
    // B[k][n] = Wk_eff[n][k]; n = l15 + 16*tile, lanes hold columns,
    // K split across lane halves symmetric to A
    v2f b0, b1, b2, b3;
    b0.x = sW[(l15 + 0) * D + k];
    b0.y = sW[(l15 + 0) * D + k + 1];
    b1.x = sW[(l15 + 16) * D + k];
    b1.y = sW[(l15 + 16) * D + k + 1];
    b2.x = sW[(l15 + 32) * D + k];
    b2.y = sW[(l15 + 32) * D + k + 1];
    b3.x = sW[(l15 + 48) * D + k];
    b3.y = sW[(l15 + 48) * D + k + 1];
    acc0 = wmma4(a, b0, acc0);
    acc1 = wmma4(a, b1, acc1);
    acc2 = wmma4(a, b2, acc2);
    acc3 = wmma4(a, b3, acc3);
  }

  // C layout: lane L holds column n=L%16 (+16*tile), rows M = 8*(L/16)+v.
  // Per-row dot with relation vector, reduced across the 16-lane half.
  float p[8];
#pragma unroll
  for (int v = 0; v < 8; ++v) {
    int er = t * 16 + 8 * h + v;
    if (er >= nkg) er = nkg - 1;
    int ir = kg_rel[er];
    const float* rv = relation_emb + (size_t)ir * D;
    float s;
    s = rv[l15 + 0] * (acc0[v] + sB[l15 + 0]);
    s += rv[l15 + 16] * (acc1[v] + sB[l15 + 16]);
    s += rv[l15 + 32] * (acc2[v] + sB[l15 + 32]);
    s += rv[l15 + 48] * (acc3[v] + sB[l15 + 48]);
    s += __shfl_xor(s, 1, 16);
    s += __shfl_xor(s, 2, 16);
    s += __shfl_xor(s, 4, 16);
    s += __shfl_xor(s, 8, 16);
    p[v] = s;
  }
  float outv = 0.0f;
#pragma unroll
  for (int v = 0; v < 8; ++v) outv = (l15 == v) ? p[v] : outv;
  if (l15 < 8) {
    int er = t * 16 + 8 * h + l15;
    if (er < nkg) {
      float a = (outv >= 0.0f) ? outv : 0.2f * outv;
      att[er] = a;
      atomicMax(segkey + kg_item[er], fkey(a));
    }
  }
}

// ---------------------------------------------------------------------------
// Phase 2: ex = exp(att - segmax); denom += ex   (att overwritten with ex)
// ---------------------------------------------------------------------------
__global__ void softmax_ex_kernel(float* __restrict__ att,
                                  const int* __restrict__ kg_item,
                                  const unsigned* __restrict__ segkey,
                                  float* __restrict__ denom, int nkg) {
  int e = blockIdx.x * blockDim.x + threadIdx.x;
  if (e >= nkg) return;
  int it = kg_item[e];
  float m = keyf(segkey[it]);
  float ex = expf(att[e] - m);
  att[e] = ex;
  atomicAdd(denom + it, ex);
}

// Phase 3: kg_agg[item] += (ex/denom[item]) * v_vec
__global__ void kg_agg_kernel(const float* __restrict__ ex,
                              const int* __restrict__ kg_item,
                              const int* __restrict__ kg_ent,
                              const float* __restrict__ entity_emb,
                              const float* __restrict__ denom,
                              float* __restrict__ kg_agg, int nkg) {
  long gid = (long)blockIdx.x * blockDim.x + threadIdx.x;
  if (gid >= (long)nkg * D) return;
  int e = (int)(gid >> 6);
  int d = (int)(gid & 63);
  int it = kg_item[e];
  float al = ex[e] / denom[it];
  float v = entity_emb[(size_t)kg_ent[e] * D + d];
  atomicAdd(kg_agg + (size_t)it * D + d, al * v);
}

// Phase 4: CF scatter: cf_item[i] += norm*user[u];  out_user[u] += norm*item[i]
__global__ void cf_scatter_kernel(const float* __restrict__ user_emb,
                                  const float* __restrict__ item_emb,
                                  const float* __restrict__ edge_norm,
                                  const int* __restrict__ cf_u,
                                  const int* __restrict__ cf_i,
                                  float* __restrict__ cf_item,
                                  float* __restrict__ out_user, int ncf) {
  long gid = (long)blockIdx.x * blockDim.x + threadIdx.x;
  if (gid >= (long)ncf * D) return;
  int e = (int)(gid >> 6);
  int d = (int)(gid & 63);
  int u = cf_u[e];
  int i = cf_i[e];
  float nrm = edge_norm[e];
  atomicAdd(cf_item + (size_t)i * D + d, nrm * user_emb[(size_t)u * D + d]);
  atomicAdd(out_user + (size_t)u * D + d, nrm * item_emb[(size_t)i * D + d]);
}

// ---------------------------------------------------------------------------
// Phase 5: gate = sigmoid(kg@Wa.T + cf@Wb.T + ba + bb); fuse. Both GEMMs
// accumulate into the same WMMA C tiles. (counts>0) == (denom>0).
// ---------------------------------------------------------------------------
__global__ void __launch_bounds__(256) gate_fuse_kernel(
    const float* __restrict__ item_emb, const float* __restrict__ kg_agg,
    const float* __restrict__ cf_item, const float* __restrict__ denom,
    const float* __restrict__ wa, const float* __restrict__ wab,
    const float* __restrict__ wb, const float* __restrict__ wbb,
    float* __restrict__ out_item, int ni, int ntiles) {
  __shared__ float sWa[D * D];
  __shared__ float sWb[D * D];
  __shared__ float sB[D];
  for (int i = threadIdx.x; i < D * D; i += blockDim.x) {
    sWa[i] = wa[i];
    sWb[i] = wb[i];
  }
  if (threadIdx.x < D) sB[threadIdx.x] = wab[threadIdx.x] + wbb[threadIdx.x];
  __syncthreads();

  const int lane = threadIdx.x & 31;
  const int wave = threadIdx.x >> 5;
  const int l15 = lane & 15;
  const int h = lane >> 4;

  const int t = blockIdx.x * 8 + wave;
  if (t >= ntiles) return;

  int rowm = t * 16 + l15;
  if (rowm >= ni) rowm = ni - 1;
  const float* ksrc =
      (denom[rowm] > 0.0f ? kg_agg : item_emb) + (size_t)rowm * D;
  const float* csrc = cf_item + (size_t)rowm * D;
  float2 ka[16], ca[16];
#pragma unroll
  for (int kk = 0; kk < 16; ++kk) {
    int k = 4 * kk + 2 * h;
    ka[kk] = *(const float2*)(ksrc + k);
    ca[kk] = *(const float2*)(csrc + k);
  }

  v8f acc0 = {}, acc1 = {}, acc2 = {}, acc3 = {};
#pragma unroll
  for (int kk = 0; kk < 16; ++kk) {
    int k = 4 * kk + 2 * h;
    v2f a1, a2;
    a1.x = ka[kk].x;
    a1.y = ka[kk].y;
    a2.x = ca[kk].x;
    a2.y = ca[kk].y;
    v2f b;
    b.x = sWa[(l15 + 0) * D + k];
    b.y = sWa[(l15 + 0) * D + k + 1];
    acc0 = wmma4(a1, b, acc0);
    b.x = sWa[(l15 + 16) * D + k];
    b.y = sWa[(l15 + 16) * D + k + 1];
    acc1 = wmma4(a1, b, acc1);
    b.x = sWa[(l15 + 32) * D + k];
    b.y = sWa[(l15 + 32) * D + k + 1];
    acc2 = wmma4(a1, b, acc2);
    b.x = sWa[(l15 + 48) * D + k];
    b.y = sWa[(l15 + 48) * D + k + 1];
    acc3 = wmma4(a1, b, acc3);
    b.x = sWb[(l15 + 0) * D + k];
    b.y = sWb[(l15 + 0) * D + k + 1];
    acc0 = wmma4(a2, b, acc0);
    b.x = sWb[(l15 + 16) * D + k];
    b.y = sWb[(l15 + 16) * D + k + 1];
    acc1 = wmma4(a2, b, acc1);
    b.x = sWb[(l15 + 32) * D + k];
    b.y = sWb[(l15 + 32) * D + k + 1];
    acc2 = wmma4(a2, b, acc2);
    b.x = sWb[(l15 + 48) * D + k];
    b.y = sWb[(l15 + 48) * D + k + 1];
    acc3 = wmma4(a2, b, acc3);
  }

#pragma unroll
  for (int v = 0; v < 8; ++v) {
    int r2 = t * 16 + 8 * h + v;
    if (r2 < ni) {
      const float* kr = (denom[r2] > 0.0f ? kg_agg : item_emb) + (size_t)r2 * D;
      const float* cr = cf_item + (size_t)r2 * D;
      float g, kv, cv;
      int n;
      n = l15 + 0;
      g = acc0[v] + sB[n];
      g = 1.0f / (1.0f + expf(-g));
      kv = kr[n];
      cv = cr[n];
      out_item[(size_t)r2 * D + n] = g * kv + (1.0f - g) * cv;
      n = l15 + 16;
      g = acc1[v] + sB[n];
      g = 1.0f / (1.0f + expf(-g));
      kv = kr[n];
      cv = cr[n];
      out_item[(size_t)r2 * D + n] = g * kv + (1.0f - g) * cv;
      n = l15 + 32;
      g = acc2[v] + sB[n];
      g = 1.0f / (1.0f + expf(-g));
      kv = kr[n];
      cv = cr[n];
      out_item[(size_t)r2 * D + n] = g * kv + (1.0f - g) * cv;
      n = l15 + 48;
      g = acc3[v] + sB[n];
      g = 1.0f / (1.0f + expf(-g));
      kv = kr[n];
      cv = cr[n];
      out_item[(size_t)r2 * D + n] = g * kv + (1.0f - g) * cv;
    }
  }
}

extern "C" void kernel_launch(void* const* d_in, const int* in_sizes, int n_in,
                              void* d_out, int out_size, void* d_ws,
                              size_t ws_size, hipStream_t stream) {
  const float* user_emb = (const float*)d_in[0];
  const float* item_emb = (const float*)d_in[1];
  const float* entity_emb = (const float*)d_in[2];
  const float* relation_emb = (const float*)d_in[3];
  const float* wkw = (const float*)d_in[4];
  const float* wkb = (const float*)d_in[5];
  const float* waw = (const float*)d_in[6];
  const float* wab = (const float*)d_in[7];
  const float* wbw = (const float*)d_in[8];
  const float* wbb = (const float*)d_in[9];
  const float* edge_norm = (const float*)d_in[10];
  const int* kg_item = (const int*)d_in[11];
  const int* kg_rel = (const int*)d_in[12];
  const int* kg_ent = (const int*)d_in[13];
  const int* cf_u = (const int*)d_in[14];
  const int* cf_i = (const int*)d_in[15];

  const int NU = in_sizes[0] / D;
  const int NI = in_sizes[1] / D;
  const int NE = in_sizes[2] / D;
  const int NCF = in_sizes[10];
  const int NKG = in_sizes[11];

  float* out_user = (float*)d_out;                 // NU*D
  float* out_item = out_user + (size_t)NU * D;     // NI*D
  float* out_ent = out_item + (size_t)NI * D;      // NE*D

  float* ws = (float*)d_ws;
  float* wkeff = ws;                                // D*D
  float* att = wkeff + D * D;                       // NKG
  unsigned* segkey = (unsigned*)(att + NKG);        // NI
  float* denom = (float*)(segkey + NI);             // NI
  float* kg_agg = denom + NI;                       // NI*D
  float* cf_item = kg_agg + (size_t)NI * D;         // NI*D

  // init: output user accumulator + [segkey | denom | kg_agg | cf_item]
  zero_f32_kernel<<<4096, 256, 0, stream>>>(out_user, (long)NU * D);
  zero_f32_kernel<<<4096, 256, 0, stream>>>((float*)segkey,
                                            (long)NI * (2 + 2 * D));
  wkeff_kernel<<<16, 256, 0, stream>>>(wkw, wkeff);

  const int ntiles = (NKG + 15) / 16;
  kg_att_kernel<<<(ntiles + 7) / 8, 256, 0, stream>>>(
      item_emb, entity_emb, relation_emb, wkeff, wkb, kg_item, kg_rel, kg_ent,
      att, segkey, NKG, ntiles);

  softmax_ex_kernel<<<(NKG + 255) / 256, 256, 0, stream>>>(att, kg_item,
                                                           segkey, denom, NKG);

  long na = (long)NKG * D;
  kg_agg_kernel<<<(unsigned)((na + 255) / 256), 256, 0, stream>>>(
      att, kg_item, kg_ent, entity_emb, denom, kg_agg, NKG);

  long nc = (long)NCF * D;
  cf_scatter_kernel<<<(unsigned)((nc + 255) / 256), 256, 0, stream>>>(
      user_emb, item_emb, edge_norm, cf_u, cf_i, cf_item, out_user, NCF);

  const int nti = (NI + 15) / 16;
  gate_fuse_kernel<<<(nti + 7) / 8, 256, 0, stream>>>(
      item_emb, kg_agg, cf_item, denom, waw, wab, wbw, wbb, out_item, NI, nti);

  copy_f32v4_kernel<<<4096, 256, 0, stream>>>(entity_emb, out_ent,
                                              (long)NE * D / 4);
}